// TrendGraph_Fusion_84198538870998
// MI455X (gfx1250) — compile-verified
//
#include <hip/hip_runtime.h>
#include <hip/hip_bf16.h>
#include <math.h>

typedef __attribute__((ext_vector_type(2))) float v2f;
typedef __attribute__((ext_vector_type(8))) float v8f;

#define BB 64
#define CC 128
#define NN 170
#define NP 176     // N padded to multiple of 16
#define TT 24
#define NTILE 11   // NP/16
#define KTOP 136   // int(0.8*170)

// ---------------------------------------------------------------------------
// Kernel 1: xs_t[b][n][c] = sum_t x[b][c][n][t], zero-padded for n in [170,176)
// Transposed + padded layout: WMMA fragments become single b64 loads, no masks.
// ---------------------------------------------------------------------------
__global__ __launch_bounds__(256) void time_sum_kernel(const float* __restrict__ x,
                                                       float* __restrict__ xs_t) {
    int j = blockIdx.x * blockDim.x + threadIdx.x;   // over B*NP*CC
    if (j >= BB * NP * CC) return;
    int c  = j % CC;
    int bn = j / CC;
    int n  = bn % NP;
    int b  = bn / NP;
    float s = 0.f;
    if (n < NN) {
        const float* p = x + (((size_t)b * CC + c) * NN + n) * TT;
#pragma unroll
        for (int t = 0; t < TT; ++t) s += p[t];
    }
    xs_t[j] = s;   // [b][n][c], coalesced
}

// ---------------------------------------------------------------------------
// Kernel 2: scores[b][n][m] = relu( (1/sqrt(C)) * sum_c xs[b][n][c]*xs[b][m][c] )
// One wave per 16x16 tile; V_WMMA_F32_16X16X4_F32 accumulated over K=C=128.
// A-fragment layout (16x4 f32): lanes 0-15 hold K=0,1; lanes 16-31 hold K=2,3.
// ---------------------------------------------------------------------------
__global__ __launch_bounds__(256) void gram_wmma_kernel(const float* __restrict__ xs_t,
                                                        float* __restrict__ scores) {
    int wave = (blockIdx.x * blockDim.x + threadIdx.x) >> 5;
    int lane = threadIdx.x & 31;

    int b  = wave / (NTILE * NTILE);
    int t  = wave % (NTILE * NTILE);
    int tn = t / NTILE;
    int tm = t % NTILE;
    int n0 = tn * 16;
    int m0 = tm * 16;

    int hi = lane >> 4;        // 0: lanes 0-15, 1: lanes 16-31
    int l  = lane & 15;
    int ka = hi * 2;           // this lane's K base within the 4-wide slab

    const float* arow = xs_t + ((size_t)b * NP + n0 + l) * CC + ka;
    const float* brow = xs_t + ((size_t)b * NP + m0 + l) * CC + ka;

    v8f acc = {};
#pragma unroll 8
    for (int c = 0; c < CC; c += 4) {
        v2f a  = *(const v2f*)(arow + c);   // global_load_b64
        v2f bf = *(const v2f*)(brow + c);   // global_load_b64
        acc = __builtin_amdgcn_wmma_f32_16x16x4_f32(
            /*neg_a=*/false, a, /*neg_b=*/false, bf,
            /*c_mod=*/(short)0, acc, /*reuse_a=*/false, /*reuse_b=*/false);
    }

    const float scale = 0.088388347648318447f;  // 1/sqrt(128)
    // C/D layout: VGPR r, lanes 0-15 -> row r, lanes 16-31 -> row r+8; col = lane&15
    float* out = scores + ((size_t)b * NP + n0 + hi * 8) * NP + m0 + l;
#pragma unroll
    for (int r = 0; r < 8; ++r) {
        float v = acc[r] * scale;
        out[(size_t)r * NP] = v > 0.f ? v : 0.f;   // unconditional, padded buffer
    }
}

// ---------------------------------------------------------------------------
// Block reductions (blockDim.x == 256)
// ---------------------------------------------------------------------------
__device__ __forceinline__ float block_max(float v, float* red) {
    int tid = threadIdx.x;
    red[tid] = v;
    __syncthreads();
    for (int s = 128; s > 0; s >>= 1) {
        if (tid < s) red[tid] = fmaxf(red[tid], red[tid + s]);
        __syncthreads();
    }
    float r = red[0];
    __syncthreads();
    return r;
}

__device__ __forceinline__ float block_sum(float v, float* red) {
    int tid = threadIdx.x;
    red[tid] = v;
    __syncthreads();
    for (int s = 128; s > 0; s >>= 1) {
        if (tid < s) red[tid] += red[tid + s];
        __syncthreads();
    }
    float r = red[0];
    __syncthreads();
    return r;
}

// ---------------------------------------------------------------------------
// Kernel 3: adj_adp = rowwise softmax(relu(E @ E^T)), one block per row n
// ---------------------------------------------------------------------------
__global__ __launch_bounds__(256) void adp_kernel(const float* __restrict__ E,
                                                  float* __restrict__ adj_adp) {
    __shared__ float red[256];
    int n   = blockIdx.x;
    int tid = threadIdx.x;

    float v = -1e30f;
    if (tid < NN) {
        float s = 0.f;
#pragma unroll
        for (int d = 0; d < 10; ++d) s += E[n * 10 + d] * E[tid * 10 + d];
        v = s > 0.f ? s : 0.f;
    }
    float m = block_max(v, red);
    float e = (tid < NN) ? __expf(v - m) : 0.f;
    float z = block_sum(e, red);
    if (tid < NN) adj_adp[(size_t)n * NN + tid] = e / z;
}

// ---------------------------------------------------------------------------
// Kernel 4: per (b,n) row: softmax(dyn) -> fuse -> softmax -> top-k mask
// ---------------------------------------------------------------------------
__global__ __launch_bounds__(256) void fuse_topk_kernel(const float* __restrict__ scores,
                                                        const float* __restrict__ adj_adp,
                                                        const float* __restrict__ fc_w,
                                                        const float* __restrict__ fc_b,
                                                        float* __restrict__ out) {
    __shared__ float red[256];
    __shared__ float frow[NN];

    int b   = blockIdx.x / NN;
    int n   = blockIdx.x % NN;
    int tid = threadIdx.x;

    const float* srow = scores + ((size_t)b * NP + n) * NP;   // padded stride

    // softmax over relu'd scores -> adj_dyn
    float v = (tid < NN) ? srow[tid] : -1e30f;
    float m = block_max(v, red);
    float e = (tid < NN) ? __expf(v - m) : 0.f;
    float z = block_sum(e, red);
    float dyn = e / z;

    // fuse: dyn*w00 + adp*w01 + bias, then softmax
    float w0 = fc_w[0], w1 = fc_w[1], fb = fc_b[0];
    float f = (tid < NN) ? (dyn * w0 + adj_adp[(size_t)n * NN + tid] * w1 + fb) : -1e30f;
    float m2 = block_max(f, red);
    float e2 = (tid < NN) ? __expf(f - m2) : 0.f;
    float z2 = block_sum(e2, red);
    float af = e2 / z2;

    if (tid < NN) frow[tid] = af;
    __syncthreads();

    // exact top-k: rank by (value desc, index asc); keep rank < KTOP
    if (tid < NN) {
        float me = frow[tid];
        int rank = 0;
        for (int j = 0; j < NN; ++j) {
            float o = frow[j];
            rank += (o > me) || (o == me && j < tid);
        }
        out[((size_t)b * NN + n) * NN + tid] = (rank < KTOP) ? me : 0.f;
    }
}

// ---------------------------------------------------------------------------
extern "C" void kernel_launch(void* const* d_in, const int* in_sizes, int n_in,
                              void* d_out, int out_size, void* d_ws, size_t ws_size,
                              hipStream_t stream) {
    const float* x    = (const float*)d_in[0];  // [64,128,170,24]
    const float* E    = (const float*)d_in[1];  // [170,10]
    const float* fc_w = (const float*)d_in[2];  // [1,2]
    const float* fc_b = (const float*)d_in[3];  // [1]
    float* out = (float*)d_out;                 // [64,170,170]

    char* ws = (char*)d_ws;
    const size_t xs_bytes     = (size_t)BB * NP * CC * sizeof(float);   // ~5.8 MB
    const size_t scores_bytes = (size_t)BB * NP * NP * sizeof(float);   // ~7.9 MB
    float* xs_t    = (float*)ws;
    float* scores  = (float*)(ws + xs_bytes);
    float* adj_adp = (float*)(ws + xs_bytes + scores_bytes);

    // 1) time reduction into transposed, zero-padded xs_t[b][n][c]
    {
        int total = BB * NP * CC;
        int grid = (total + 255) / 256;
        time_sum_kernel<<<grid, 256, 0, stream>>>(x, xs_t);
    }
    // 2) batched gram via WMMA (7744 waves, 8 waves/block -> 968 blocks exactly)
    {
        int waves = BB * NTILE * NTILE;
        int grid = waves / 8;
        gram_wmma_kernel<<<grid, 256, 0, stream>>>(xs_t, scores);
    }
    // 3) adaptive adjacency (batch invariant)
    adp_kernel<<<NN, 256, 0, stream>>>(E, adj_adp);
    // 4) fuse + softmax + top-k mask, one block per (b,n) row
    fuse_topk_kernel<<<BB * NN, 256, 0, stream>>>(scores, adj_adp, fc_w, fc_b, out);
}